// fagcn_45423574123070
// MI455X (gfx1250) — compile-verified
//
#include <hip/hip_runtime.h>
#include <math.h>

typedef __attribute__((ext_vector_type(2))) float v2f;
typedef __attribute__((ext_vector_type(8))) float v8f;

#define FA_EPS 0.5f
#define DIM 64

// ---------------------------------------------------------------------------
// Small utility kernels
// ---------------------------------------------------------------------------
__global__ void fa_zero_f32(float* __restrict__ p, int n) {
    int i = blockIdx.x * blockDim.x + threadIdx.x;
    if (i < n) p[i] = 0.0f;
}

__global__ void fa_deg_accum(const int* __restrict__ dst, float* __restrict__ deg, int E) {
    int e = blockIdx.x * blockDim.x + threadIdx.x;
    if (e < E) atomicAdd(&deg[dst[e]], 1.0f);
}

__global__ void fa_dinv_finalize(float* __restrict__ deg, int n) {
    int i = blockIdx.x * blockDim.x + threadIdx.x;
    if (i < n) {
        float d = deg[i];
        deg[i] = (d > 0.0f) ? rsqrtf(d) : 0.0f;
    }
}

// out = FA_EPS * x0   (vectorized float4; n4 = N*DIM/4)
__global__ void fa_scale_init(float4* __restrict__ out, const float4* __restrict__ x0, int n4) {
    int i = blockIdx.x * blockDim.x + threadIdx.x;
    if (i < n4) {
        float4 v = x0[i];
        v.x *= FA_EPS; v.y *= FA_EPS; v.z *= FA_EPS; v.w *= FA_EPS;
        out[i] = v;
    }
}

// ---------------------------------------------------------------------------
// WMMA fp32 matvec-x2:  y = X[N,64] @ B[64,16]  with B col0=w0, col1=w1, rest 0.
// One wave per 16-row tile; 16 x V_WMMA_F32_16X16X4_F32 over K=64.
// Branch-free B construction: per-lane pointer select (loop-invariant
// v_cndmask on the pointer) + multiplicative zero mask for cols >= 2, so the
// unrolled body is just 2x global_load_b64 + 2x v_mul + v_wmma.
// A 16x4 layout: lanes 0-15 hold (K=k0,k0+1), lanes 16-31 hold (K=k0+2,k0+3).
// B 4x16 layout: same half-wave split; col = lane & 15.
// C/D 16x16:  VGPR v -> row v (lanes 0-15) / row v+8 (lanes 16-31).
// ---------------------------------------------------------------------------
__global__ void fa_matvec2_wmma(const float* __restrict__ X, int nrows,
                                const float* __restrict__ w0,
                                const float* __restrict__ w1,
                                const float* __restrict__ bias, // 2 elems or nullptr
                                float* __restrict__ out0, int s0,
                                float* __restrict__ out1, int s1) {
    const int lane    = threadIdx.x & 31;
    const int waveId  = threadIdx.x >> 5;
    const int tile    = blockIdx.x * (blockDim.x >> 5) + waveId;
    const int rowBase = tile * 16;
    if (rowBase >= nrows) return;  // wave-uniform: EXEC stays all-ones for WMMA

    const int half = lane >> 4;        // 0 or 1
    const int col  = lane & 15;
    const int kb   = half * 2;         // K sub-offset within the 4-wide slab

    int row = rowBase + (lane & 15);
    if (row >= nrows) row = nrows - 1; // clamp loads only; stores are guarded

    const float* __restrict__ xrow = X + (long)row * DIM;
    // loop-invariant per-lane select: which weight column this lane supplies
    const float* __restrict__ wsel = (col == 0) ? w0 : w1;
    const float  bmask = (col < 2) ? 1.0f : 0.0f;

    v8f acc = {};
#pragma unroll
    for (int k0 = 0; k0 < DIM; k0 += 4) {
        const int kk = k0 + kb;
        v2f a = *reinterpret_cast<const v2f*>(xrow + kk);  // global_load_b64
        v2f w = *reinterpret_cast<const v2f*>(wsel + kk);  // global_load_b64 (L0-hot)
        v2f b;
        b.x = bmask * w.x;
        b.y = bmask * w.y;
        acc = __builtin_amdgcn_wmma_f32_16x16x4_f32(
            /*neg_a=*/false, a, /*neg_b=*/false, b,
            /*c_mod=*/(short)0, acc, /*reuse_a=*/false, /*reuse_b=*/false);
    }

    const float b0 = bias ? bias[0] : 0.0f;
    const float b1 = bias ? bias[1] : 0.0f;
#pragma unroll
    for (int v = 0; v < 8; ++v) {
        const int r = rowBase + half * 8 + v;
        if (r < nrows) {
            if (col == 0)      out0[(long)r * s0] = acc[v] + b0;
            else if (col == 1) out1[(long)r * s1] = acc[v] + b1;
        }
    }
}

// ---------------------------------------------------------------------------
// Edge scatter: 16 lanes per edge, each lane does a float4 gather
// (global_load_b128) and 4 global_atomic_add_f32 scatters (no-return ->
// STOREcnt, no wave stall). x / out (25.6MB each) are L2-resident on MI455X.
// ---------------------------------------------------------------------------
__global__ void fa_edge_scatter(const float* __restrict__ xcur,
                                const int* __restrict__ src,
                                const int* __restrict__ dst,
                                const float* __restrict__ xl,
                                const float* __restrict__ xr,
                                const float* __restrict__ dinv,
                                float* __restrict__ out, int E) {
    const int t = blockIdx.x * blockDim.x + threadIdx.x;
    const int e = t >> 4;
    if (e >= E) return;
    const int q = (t & 15) << 2;  // dim offset 0,4,...,60

    const int s = src[e];
    const int d = dst[e];
    const float coef = tanhf(xl[s] + xr[d]) * dinv[s] * dinv[d];

    const float4 v = *reinterpret_cast<const float4*>(xcur + (long)s * DIM + q);
    float* o = out + (long)d * DIM + q;
    atomicAdd(o + 0, coef * v.x);
    atomicAdd(o + 1, coef * v.y);
    atomicAdd(o + 2, coef * v.z);
    atomicAdd(o + 3, coef * v.w);
}

// ---------------------------------------------------------------------------
// Host orchestration
// ---------------------------------------------------------------------------
extern "C" void kernel_launch(void* const* d_in, const int* in_sizes, int n_in,
                              void* d_out, int out_size, void* d_ws, size_t ws_size,
                              hipStream_t stream) {
    const float* x       = (const float*)d_in[0];      // [N,64]
    const int*   edges   = (const int*)d_in[1];        // [2,E]
    const float* att_l   = (const float*)d_in[2];      // [64]
    const float* att_r   = (const float*)d_in[3];      // [64]
    const float* pred_w  = (const float*)d_in[4];      // [2,64]
    const float* pred_b  = (const float*)d_in[5];      // [2]
    float* out = (float*)d_out;                        // [N,2]

    const int N = in_sizes[0] / DIM;
    const int E = in_sizes[1] / 2;
    const int* src = edges;
    const int* dst = edges + E;

    // workspace layout
    float* dinv = (float*)d_ws;        // [N]
    float* xl   = dinv + N;            // [N]
    float* xr   = xl + N;              // [N]
    float* h1   = xr + N;              // [N*64]
    float* h2   = h1 + (long)N * DIM;  // [N*64]

    const int TB = 256;
    const int gN   = (N + TB - 1) / TB;
    const int gE   = (E + TB - 1) / TB;
    const int n4   = N * DIM / 4;
    const int gN4  = (n4 + TB - 1) / TB;
    const int gSc  = (E * 16 + TB - 1) / TB;
    const int tiles  = (N + 15) / 16;
    const int gWmma  = (tiles + 7) / 8;  // 8 waves (tiles) per 256-thread block

    // degree -> dinv (shared by both layers)
    fa_zero_f32<<<gN, TB, 0, stream>>>(dinv, N);
    fa_deg_accum<<<gE, TB, 0, stream>>>(dst, dinv, E);
    fa_dinv_finalize<<<gN, TB, 0, stream>>>(dinv, N);

    // ---- layer 1: h1 = scatter(x) + EPS*x ----
    fa_matvec2_wmma<<<gWmma, TB, 0, stream>>>(x, N, att_l, att_r, nullptr, xl, 1, xr, 1);
    fa_scale_init<<<gN4, TB, 0, stream>>>((float4*)h1, (const float4*)x, n4);
    fa_edge_scatter<<<gSc, TB, 0, stream>>>(x, src, dst, xl, xr, dinv, h1, E);

    // ---- layer 2: h2 = scatter(h1) + EPS*x ----
    fa_matvec2_wmma<<<gWmma, TB, 0, stream>>>(h1, N, att_l, att_r, nullptr, xl, 1, xr, 1);
    fa_scale_init<<<gN4, TB, 0, stream>>>((float4*)h2, (const float4*)x, n4);
    fa_edge_scatter<<<gSc, TB, 0, stream>>>(h1, src, dst, xl, xr, dinv, h2, E);

    // ---- prediction head: out = h2 @ pred_w.T + pred_b ----
    fa_matvec2_wmma<<<gWmma, TB, 0, stream>>>(h2, N, pred_w, pred_w + DIM, pred_b,
                                              out, 2, out + 1, 2);
}